// Projection_61787399520788
// MI455X (gfx1250) — compile-verified
//
#include <hip/hip_runtime.h>
#include <hip/hip_bf16.h>

// ---------------------------------------------------------------------------
// 8x8 complex expm via real 16x16 embedding + V_WMMA_F32_16X16X4_F32 chains.
// One wave32 per (n, channel) matrix. 16x16x16 matmul = 4 chained WMMAs.
// Cross-half lane exchange done with v_permlanex16_b32 (pure VALU, no DS).
// ---------------------------------------------------------------------------

typedef __attribute__((ext_vector_type(2))) float v2f;
typedef __attribute__((ext_vector_type(8))) float v8f;

// lane <- opposite 16-lane row, same position (i.e. lane ^ 16), pure VALU
__device__ __forceinline__ float xswap16(float x) {
    int i = __float_as_int(x);
    return __int_as_float(__builtin_amdgcn_permlanex16(
        i, i, 0x76543210, 0xfedcba98, false, false));
}

__device__ __forceinline__ float swz_xor1(float x)  { return __int_as_float(__builtin_amdgcn_ds_swizzle(__float_as_int(x), 0x041F)); }
__device__ __forceinline__ float swz_xor2(float x)  { return __int_as_float(__builtin_amdgcn_ds_swizzle(__float_as_int(x), 0x081F)); }
__device__ __forceinline__ float swz_xor4(float x)  { return __int_as_float(__builtin_amdgcn_ds_swizzle(__float_as_int(x), 0x101F)); }
__device__ __forceinline__ float swz_xor8(float x)  { return __int_as_float(__builtin_amdgcn_ds_swizzle(__float_as_int(x), 0x201F)); }

__device__ __forceinline__ v8f wmma4(v2f a, v2f b, v8f c) {
    // D = A(16x4) x B(4x16) + C(16x16), f32
    return __builtin_amdgcn_wmma_f32_16x16x4_f32(false, a, false, b, (short)0, c,
                                                 false, false);
}

// Full 16x16x16 product:  D = As(A-layout) @ T(C-layout).
// All 8 half-swaps are computed unconditionally (VALU permlanex16), then the
// B-operand sub-blocks are formed with v_cndmask selects, then 4 WMMAs chain.
__device__ __forceinline__ v8f mat16_mul(const v2f a[4], const float t[8], bool hi) {
    float ts[8];
#pragma unroll
    for (int v = 0; v < 8; ++v) ts[v] = xswap16(t[v]);
    v2f b0, b1, b2, b3;
    b0.x = hi ? ts[2] : t[0];   b0.y = hi ? ts[3] : t[1];   // rows 0..3
    b1.x = hi ? ts[6] : t[4];   b1.y = hi ? ts[7] : t[5];   // rows 4..7
    b2.x = hi ? t[2]  : ts[0];  b2.y = hi ? t[3]  : ts[1];  // rows 8..11
    b3.x = hi ? t[6]  : ts[4];  b3.y = hi ? t[7]  : ts[5];  // rows 12..15
    v8f acc = {};
    acc = wmma4(a[0], b0, acc);
    acc = wmma4(a[1], b1, acc);
    acc = wmma4(a[2], b2, acc);
    acc = wmma4(a[3], b3, acc);
    return acc;
}

// ---------------------------------------------------------------------------
// Pre-pass: embed Ask[i,c] (skew-hermitian projection of A) into the real
// 16x16 matrix [[R,-I],[I,R]], stored as ws[(c*256 + e)*16 + i] so the main
// kernel's contraction over i is a contiguous float4 x4 load.
// ---------------------------------------------------------------------------
__global__ __launch_bounds__(256) void embed_kernel(const float* __restrict__ Af,
                                                    float* __restrict__ ws) {
    int flat = blockIdx.x * 256 + threadIdx.x;   // 0 .. 32767
    int i = flat & 15;
    int e = (flat >> 4) & 255;
    int c = flat >> 12;
    int r = e >> 4, k = e & 15;
    int h = r & 7, kk = k & 7;
    int base = (i * 8 + c) * 8;
    float ar_hk = Af[2 * ((base + h) * 8 + kk)    ];
    float ai_hk = Af[2 * ((base + h) * 8 + kk) + 1];
    float ar_kh = Af[2 * ((base + kk) * 8 + h)    ];
    float ai_kh = Af[2 * ((base + kk) * 8 + h) + 1];
    float re = 0.5f * (ar_hk - ar_kh);   // Ask = 0.5*(A - A^H)
    float im = 0.5f * (ai_hk + ai_kh);
    float val;
    if (r < 8) val = (k < 8) ?  re : -im;
    else       val = (k < 8) ?  im :  re;
    ws[flat] = val;
}

// ---------------------------------------------------------------------------
// Main kernel: block = one n (8 waves = 8 channels). Each wave owns one 16x16
// real matrix spread in WMMA C-layout across 8 VGPRs.
// ---------------------------------------------------------------------------
__global__ __launch_bounds__(256) void expm_kernel(const float* __restrict__ dX,
                                                   const float* __restrict__ emb,
                                                   float* __restrict__ out) {
    __shared__ float lds[8 * 256];               // 1KB per wave transpose slab
    const int tid = threadIdx.x;
    const int l   = tid & 31;
    const int w   = tid >> 5;                    // channel
    const int n   = blockIdx.x;
    const bool hi = (l & 16) != 0;
    const int col = l & 15;
    const int rowoff = hi ? 8 : 0;
    float* ldsw = lds + w * 256;

    // dX[n, 0..15] (wave-uniform)
    float dxv[16];
    const float4* dxp = (const float4*)(dX + (size_t)n * 16);
#pragma unroll
    for (int q = 0; q < 4; ++q) {
        float4 t = dxp[q];
        dxv[4*q] = t.x; dxv[4*q+1] = t.y; dxv[4*q+2] = t.z; dxv[4*q+3] = t.w;
    }

    // M = sum_i dX[i] * emb[i][c]  accumulated directly in C-layout
    float m[8];
#pragma unroll
    for (int v = 0; v < 8; ++v) {
        const float4* wp = (const float4*)(emb +
            (((size_t)w * 256 + (size_t)(v + rowoff) * 16 + col) << 4));
        float4 q0 = wp[0], q1 = wp[1], q2 = wp[2], q3 = wp[3];
        float a = 0.f;
        a = fmaf(dxv[0],  q0.x, a); a = fmaf(dxv[1],  q0.y, a);
        a = fmaf(dxv[2],  q0.z, a); a = fmaf(dxv[3],  q0.w, a);
        a = fmaf(dxv[4],  q1.x, a); a = fmaf(dxv[5],  q1.y, a);
        a = fmaf(dxv[6],  q1.z, a); a = fmaf(dxv[7],  q1.w, a);
        a = fmaf(dxv[8],  q2.x, a); a = fmaf(dxv[9],  q2.y, a);
        a = fmaf(dxv[10], q2.z, a); a = fmaf(dxv[11], q2.w, a);
        a = fmaf(dxv[12], q3.x, a); a = fmaf(dxv[13], q3.y, a);
        a = fmaf(dxv[14], q3.z, a); a = fmaf(dxv[15], q3.w, a);
        m[v] = a;
    }

    // complex induced 1-norm: Re / Im of a column sit in opposite lane halves
    float cs = 0.f;
#pragma unroll
    for (int v = 0; v < 8; ++v) {
        float p = xswap16(m[v]);
        cs += sqrtf(fmaf(m[v], m[v], p * p));
    }
    cs = fmaxf(cs, swz_xor1(cs));
    cs = fmaxf(cs, swz_xor2(cs));
    cs = fmaxf(cs, swz_xor4(cs));
    cs = fmaxf(cs, swz_xor8(cs));
    cs = fmaxf(cs, xswap16(cs));
    float lg    = ceilf(log2f(fmaxf(cs, 1.0f)));
    float sf    = fminf(fmaxf(lg, 0.0f), 16.0f);
    int   si    = __builtin_amdgcn_readfirstlane((int)sf);   // wave-uniform
    float scale = exp2f(-sf);
#pragma unroll
    for (int v = 0; v < 8; ++v) m[v] *= scale;

    // A-operand layout of As via 1KB/wave LDS transpose (built once)
    v2f aA[4];
    {
#pragma unroll
        for (int v = 0; v < 8; ++v) ldsw[(v + rowoff) * 16 + col] = m[v];
        asm volatile("s_wait_dscnt 0" ::: "memory");
        int row = l & 15, h2 = hi ? 2 : 0;
#pragma unroll
        for (int j = 0; j < 4; ++j) {
            aA[j].x = ldsw[row * 16 + 4 * j + h2];
            aA[j].y = ldsw[row * 16 + 4 * j + 1 + h2];
        }
        asm volatile("s_wait_dscnt 0" ::: "memory");
    }

    // Taylor: E = I + As + ... ; term_k = As @ term_{k-1} / k  (As commutes)
    float term[8], E[8];
#pragma unroll
    for (int v = 0; v < 8; ++v) {
        term[v] = m[v];
        E[v]    = m[v] + (((v + rowoff) == col) ? 1.0f : 0.0f);
    }
    const float invk[19] = {0.f, 1.f, 1.f/2.f, 1.f/3.f, 1.f/4.f, 1.f/5.f,
                            1.f/6.f, 1.f/7.f, 1.f/8.f, 1.f/9.f, 1.f/10.f,
                            1.f/11.f, 1.f/12.f, 1.f/13.f, 1.f/14.f, 1.f/15.f,
                            1.f/16.f, 1.f/17.f, 1.f/18.f};
#pragma unroll
    for (int k = 2; k <= 18; ++k) {
        v8f p = mat16_mul(aA, term, hi);
#pragma unroll
        for (int v = 0; v < 8; ++v) {
            term[v] = p[v] * invk[k];
            E[v]   += term[v];
        }
    }

    // repeated squaring, s times (uniform scalar loop)
    for (int it = 0; it < si; ++it) {
        v2f aE[4];
#pragma unroll
        for (int v = 0; v < 8; ++v) ldsw[(v + rowoff) * 16 + col] = E[v];
        asm volatile("s_wait_dscnt 0" ::: "memory");
        int row = l & 15, h2 = hi ? 2 : 0;
#pragma unroll
        for (int j = 0; j < 4; ++j) {
            aE[j].x = ldsw[row * 16 + 4 * j + h2];
            aE[j].y = ldsw[row * 16 + 4 * j + 1 + h2];
        }
        asm volatile("s_wait_dscnt 0" ::: "memory");
        v8f p = mat16_mul(aE, E, hi);
#pragma unroll
        for (int v = 0; v < 8; ++v) E[v] = p[v];
    }

    // store complex64: Re = top-left quadrant (lane-lo), Im = bottom-left
    // (lane-hi). Lanes 0..7 write (re,im) pairs -> contiguous 64B segments.
    float im8[8];
#pragma unroll
    for (int v = 0; v < 8; ++v) im8[v] = xswap16(E[v]);
    if (l < 8) {
        float2* ob = (float2*)(out + ((size_t)n * 8 + w) * 128);
#pragma unroll
        for (int v = 0; v < 8; ++v) ob[v * 8 + l] = make_float2(E[v], im8[v]);
    }
}

extern "C" void kernel_launch(void* const* d_in, const int* in_sizes, int n_in,
                              void* d_out, int out_size, void* d_ws, size_t ws_size,
                              hipStream_t stream) {
    const float* dX = (const float*)d_in[0];          // (N,16) f32
    const float* A  = (const float*)d_in[1];          // (16,8,8,8) complex64 interleaved
    float* out = (float*)d_out;                       // (N,8,8,8) complex64 interleaved
    float* emb = (float*)d_ws;                        // 32768 f32 = 128KB scratch

    const int N = in_sizes[0] / 16;                   // 65536

    embed_kernel<<<128, 256, 0, stream>>>(A, emb);
    expm_kernel<<<N, 256, 0, stream>>>(dX, emb, out);
}